// LocallyConnected2d_12292196401715
// MI455X (gfx1250) — compile-verified
//
#include <hip/hip_runtime.h>

typedef __attribute__((ext_vector_type(2))) float v2f;
typedef __attribute__((ext_vector_type(8))) float v8f;

#define KSZ 7
#define KK 49            // real taps
#define PADK 3
#define Bn 16
#define Cn 4
#define Hn 128
#define Wn 128

#define TILE_W 8                     // pixels per workgroup (one wave each)
#define HALO_W (TILE_W + KSZ - 1)    // 14
#define XPLANE (Cn * KSZ * HALO_W)   // 392 floats per batch slice
#define XSTRIDE (XPLANE + 1)         // 393: lane stride 9 mod 64 banks (conflict-free)

// Re-ordered, padded contraction: k = (i*7 + dy)*8 + dx, dx in 0..7 (dx==7 -> weight 0)
#define NROWS (Cn * KSZ)             // 28 (i,dy) rows
#define KPAD (NROWS * 8)             // 224 padded K
#define WN_OSTRIDE 228               // per-o stride: even (b64 align), 228%64=36 bank spread
#define WN_PIX (Cn * WN_OSTRIDE)     // 912 floats per pixel, layout [o][k]

__global__ __launch_bounds__(256)
void lc2d_wmma_kernel(const float* __restrict__ x,
                      const float* __restrict__ weight,
                      float* __restrict__ out)
{
    __shared__ float x_lds[Bn * XSTRIDE];       // 6288 floats
    __shared__ float wn_lds[TILE_W * WN_PIX];   // 7296 floats
    __shared__ float zclear[KPAD];              // zero B-column for N>=4 lanes

    const unsigned tid  = threadIdx.x;
    const unsigned bid  = blockIdx.x;           // 2048 blocks
    const unsigned h    = bid >> 4;
    const unsigned w0   = (bid & 15u) << 3;
    const unsigned wave = tid >> 5;
    const unsigned lane = tid & 31u;
    const unsigned w    = w0 + wave;

    // ---- stage x halo tile + zero column ----
    if (tid < KPAD) zclear[tid] = 0.0f;
    for (unsigned idx = tid; idx < Bn * XPLANE; idx += 256u) {
        unsigned b   = idx / XPLANE;
        unsigned rem = idx - b * XPLANE;
        unsigned ci  = rem / (KSZ * HALO_W);
        unsigned r2  = rem - ci * (KSZ * HALO_W);
        unsigned r   = r2 / HALO_W;
        unsigned cc  = r2 - r * HALO_W;
        int gh = (int)h  - PADK + (int)r;
        int gw = (int)w0 - PADK + (int)cc;
        float v = 0.0f;
        if (gh >= 0 && gh < Hn && gw >= 0 && gw < Wn)
            v = x[(((b * Cn + ci) * Hn + (unsigned)gh) * Wn) + (unsigned)gw];
        x_lds[b * XSTRIDE + rem] = v;
    }

    // ---- per wave: L1-normalize |weight + delta_center|, store as [o][k_pad] ----
    if (lane < 16u) {
        unsigned o = lane >> 2;
        unsigned i = lane & 3u;
        unsigned wbase = (((o * Cn + i) * Hn + h) * Wn + w) * KK;
        float wreg[KK];
        float ssum = 0.0f;
        #pragma unroll
        for (int t = 0; t < KK; ++t) {
            float wv = weight[wbase + (unsigned)t];
            if (t == KK / 2) wv += 1.0f;          // delta kernel at center tap
            wv = fabsf(wv);
            wreg[t] = wv;
            ssum += wv;
        }
        float inv = 1.0f / ssum;
        unsigned pb = wave * WN_PIX + o * WN_OSTRIDE;
        #pragma unroll
        for (int t = 0; t < KK; ++t) {
            int dy = t / KSZ, dx = t - dy * KSZ;             // compile-time
            wn_lds[pb + (unsigned)((i * KSZ + dy) * 8 + dx)] = wreg[t] * inv;
        }
        #pragma unroll
        for (int dy = 0; dy < KSZ; ++dy)                     // dx==7 pad taps
            wn_lds[pb + (unsigned)((i * KSZ + dy) * 8 + 7)] = 0.0f;
    }

    __syncthreads();

    // ---- 16 x 224 x 4 per-pixel GEMM: 56 chained V_WMMA_F32_16X16X4_F32 ----
    const unsigned hf   = lane >> 4;      // K-pair: lanes 0-15 -> {0,1}, 16-31 -> {2,3}
    const unsigned nb   = lane & 15u;     // A: batch (M) / B: out-channel col (N)
    const unsigned colb = w - w0;

    // All inner-loop addresses are immediates off these two per-lane bases.
    const float* arow = &x_lds[nb * XSTRIDE + colb + hf * 2u];
    const float* brow = (nb < (unsigned)Cn)
                      ? &wn_lds[wave * WN_PIX + nb * WN_OSTRIDE + hf * 2u]
                      : &zclear[hf * 2u];

    v8f acc = {};
    #pragma unroll
    for (int r = 0; r < NROWS; ++r) {
        #pragma unroll
        for (int ch = 0; ch < 2; ++ch) {
            v2f a;
            a[0] = arow[r * HALO_W + ch * 4 + 0];
            a[1] = arow[r * HALO_W + ch * 4 + 1];
            v2f bv = *(const v2f*)&brow[r * 8 + ch * 4];   // ds_load_b64
            acc = __builtin_amdgcn_wmma_f32_16x16x4_f32(
                      false, a, false, bv, (short)0, acc, false, false);
        }
    }

    // ---- masked store: lane (N=o), VGPR v (M = v + hf*8) ----
    if (nb < (unsigned)Cn) {
        #pragma unroll
        for (int v = 0; v < 8; ++v) {
            unsigned b  = (unsigned)v + hf * 8u;
            float xc = x_lds[b * XSTRIDE + (nb * KSZ + PADK) * HALO_W + colb + PADK];
            float ov = (xc != 0.0f) ? acc[v] : 0.0f;
            out[((b * Cn + nb) * Hn + h) * Wn + w] = ov;
        }
    }
}

extern "C" void kernel_launch(void* const* d_in, const int* in_sizes, int n_in,
                              void* d_out, int out_size, void* d_ws, size_t ws_size,
                              hipStream_t stream) {
    const float* x      = (const float*)d_in[0];
    const float* weight = (const float*)d_in[1];
    float* out          = (float*)d_out;
    dim3 grid((Hn * Wn) / TILE_W);   // 2048 workgroups, 8 waves / 8 pixels each
    dim3 block(256);
    hipLaunchKernelGGL(lc2d_wmma_kernel, grid, block, 0, stream, x, weight, out);
}